// GCN_8117488189465
// MI455X (gfx1250) — compile-verified
//
#include <hip/hip_runtime.h>
#include <math.h>

typedef float v2f __attribute__((ext_vector_type(2)));
typedef float v8f __attribute__((ext_vector_type(8)));

#define F_IN  128
#define H_OUT 25
#define H_PAD 32
#define C_OUT 40
#define C_PAD 48

__device__ __forceinline__ void atomAddF(float* p, float v) {
    // native global_atomic_add_f32 (no CAS loop)
    unsafeAtomicAdd(p, v);
}

#define WMMA_F32(A, B, C) \
    __builtin_amdgcn_wmma_f32_16x16x4_f32(false, (A), false, (B), (short)0, (C), false, false)

// ---------------- normalization ----------------
__global__ void k_init_deg(float* deg, int N) {
    int i = blockIdx.x * blockDim.x + threadIdx.x;
    if (i < N) deg[i] = 1.0f;              // self-loop weight
}

__global__ void k_deg_scatter(const int* __restrict__ dst, const float* __restrict__ w,
                              float* deg, int E) {
    int e = blockIdx.x * blockDim.x + threadIdx.x;
    if (e < E) atomAddF(&deg[dst[e]], w[e]);
}

__global__ void k_rsqrt(float* deg, int N) {
    int i = blockIdx.x * blockDim.x + threadIdx.x;
    if (i < N) { float d = deg[i]; deg[i] = (d > 0.0f) ? rsqrtf(d) : 0.0f; }
}

__global__ void k_norm(const int* __restrict__ src, const int* __restrict__ dst,
                       const float* __restrict__ w, const float* __restrict__ dinv,
                       float* __restrict__ norm, int E) {
    int e = blockIdx.x * blockDim.x + threadIdx.x;
    if (e < E) norm[e] = dinv[src[e]] * w[e] * dinv[dst[e]];
}

// ---------------- GEMM 1: h1 = x @ W1  (K=128, Nout padded 25->32) ----------------
// LDS holds W1 "fragment-ready": Wl[p*H_PAD+n] = {W1[2p][n], W1[2p+1][n]} -> one ds_load_b64/frag.
// Each wave computes a 32x32 output tile: 2 A-frags + 2 B-frags feed 4 WMMAs per k-step.
__global__ void __launch_bounds__(128)
k_gemm1(const float* __restrict__ x, const float* __restrict__ W1,
        float* __restrict__ h1, int N) {
    __shared__ v2f Wl[(F_IN / 2) * H_PAD];
    for (int i = threadIdx.x; i < (F_IN / 2) * H_PAD; i += 128) {
        int p = i >> 5, n = i & 31;
        v2f w;
        w.x = (n < H_OUT) ? W1[(2 * p)     * H_OUT + n] : 0.0f;
        w.y = (n < H_OUT) ? W1[(2 * p + 1) * H_OUT + n] : 0.0f;
        Wl[i] = w;
    }
    __syncthreads();

    int lane = threadIdx.x & 31;
    int wv   = threadIdx.x >> 5;
    int m0   = blockIdx.x * 128 + wv * 32;
    if (m0 >= N) return;                     // N % 32 == 0 -> no partial tiles

    int m  = lane & 15;
    int hp = lane >> 4;                      // half-wave select (K-split)
    int n  = lane & 15;
    const float* arow0 = x + (size_t)(m0 + m) * F_IN + (hp << 1);
    const float* arow1 = arow0 + (size_t)16 * F_IN;

    v8f acc00 = {}, acc01 = {}, acc10 = {}, acc11 = {};
    for (int k0 = 0; k0 < F_IN; k0 += 4) {
        v2f a0 = *(const v2f*)(arow0 + k0);
        v2f a1 = *(const v2f*)(arow1 + k0);
        int p  = (k0 >> 1) + hp;
        v2f b0 = Wl[p * H_PAD + n];
        v2f b1 = Wl[p * H_PAD + 16 + n];
        acc00 = WMMA_F32(a0, b0, acc00);
        acc01 = WMMA_F32(a0, b1, acc01);
        acc10 = WMMA_F32(a1, b0, acc10);
        acc11 = WMMA_F32(a1, b1, acc11);
    }
    int rbase = m0 + (hp << 3);
    #pragma unroll
    for (int r = 0; r < 8; ++r) {
        h1[(size_t)(rbase + r)      * H_PAD + n]      = acc00[r];
        h1[(size_t)(rbase + r)      * H_PAD + 16 + n] = acc01[r];
        h1[(size_t)(rbase + 16 + r) * H_PAD + n]      = acc10[r];
        h1[(size_t)(rbase + 16 + r) * H_PAD + 16 + n] = acc11[r];
    }
}

// ---------------- GEMM 2: h2 = h1r @ W2  (K padded 25->32, Nout padded 40->48) ----------------
__global__ void __launch_bounds__(128)
k_gemm2(const float* __restrict__ h1r, const float* __restrict__ W2,
        float* __restrict__ h2, int N) {
    __shared__ v2f Wl[(H_PAD / 2) * C_PAD];
    for (int i = threadIdx.x; i < (H_PAD / 2) * C_PAD; i += 128) {
        int p = i / C_PAD, n = i % C_PAD;
        v2f w;
        w.x = (2 * p     < H_OUT && n < C_OUT) ? W2[(2 * p)     * C_OUT + n] : 0.0f;
        w.y = (2 * p + 1 < H_OUT && n < C_OUT) ? W2[(2 * p + 1) * C_OUT + n] : 0.0f;
        Wl[i] = w;
    }
    __syncthreads();

    int lane = threadIdx.x & 31;
    int wv   = threadIdx.x >> 5;
    int m0   = blockIdx.x * 128 + wv * 32;
    if (m0 >= N) return;

    int m  = lane & 15;
    int hp = lane >> 4;
    int n  = lane & 15;
    const float* arow0 = h1r + (size_t)(m0 + m) * H_PAD + (hp << 1);
    const float* arow1 = arow0 + (size_t)16 * H_PAD;

    v8f acc00 = {}, acc01 = {}, acc02 = {};
    v8f acc10 = {}, acc11 = {}, acc12 = {};
    for (int k0 = 0; k0 < H_PAD; k0 += 4) {
        v2f a0 = *(const v2f*)(arow0 + k0);
        v2f a1 = *(const v2f*)(arow1 + k0);
        int p  = (k0 >> 1) + hp;
        v2f b0 = Wl[p * C_PAD + n];
        v2f b1 = Wl[p * C_PAD + 16 + n];
        v2f b2 = Wl[p * C_PAD + 32 + n];
        acc00 = WMMA_F32(a0, b0, acc00);
        acc01 = WMMA_F32(a0, b1, acc01);
        acc02 = WMMA_F32(a0, b2, acc02);
        acc10 = WMMA_F32(a1, b0, acc10);
        acc11 = WMMA_F32(a1, b1, acc11);
        acc12 = WMMA_F32(a1, b2, acc12);
    }
    int rbase = m0 + (hp << 3);
    #pragma unroll
    for (int r = 0; r < 8; ++r) {
        h2[(size_t)(rbase + r)      * C_PAD + n]      = acc00[r];
        h2[(size_t)(rbase + r)      * C_PAD + 16 + n] = acc01[r];
        h2[(size_t)(rbase + r)      * C_PAD + 32 + n] = acc02[r];
        h2[(size_t)(rbase + 16 + r) * C_PAD + n]      = acc10[r];
        h2[(size_t)(rbase + 16 + r) * C_PAD + 16 + n] = acc11[r];
        h2[(size_t)(rbase + 16 + r) * C_PAD + 32 + n] = acc12[r];
    }
}

// ---------------- edge aggregation ----------------
// agg1 init with self-loop message: agg1 = dinv^2 * h1 (padding cols are 0 in h1)
__global__ void k_init_agg1(const float* __restrict__ h1, const float* __restrict__ dinv,
                            float* __restrict__ agg1, long long total) {
    long long idx = (long long)blockIdx.x * blockDim.x + threadIdx.x;
    if (idx >= total) return;
    int node = (int)(idx >> 5);
    float d = dinv[node];
    agg1[idx] = d * d * h1[idx];
}

// 32 lanes per edge (f = idx&31, mask f<25): e is wave-uniform, atomics contiguous per edge
__global__ void k_scatter1(const float* __restrict__ h1, const int* __restrict__ src,
                           const int* __restrict__ dst, const float* __restrict__ norm,
                           float* agg1, int E) {
    long long idx = (long long)blockIdx.x * blockDim.x + threadIdx.x;
    int e = (int)(idx >> 5);
    int f = (int)(idx & 31);
    if (e >= E || f >= H_OUT) return;
    float v = norm[e] * h1[(size_t)src[e] * H_PAD + f];
    atomAddF(&agg1[(size_t)dst[e] * H_PAD + f], v);
}

__global__ void k_bias_relu(float* agg1, const float* __restrict__ b1, long long total) {
    long long idx = (long long)blockIdx.x * blockDim.x + threadIdx.x;
    if (idx >= total) return;
    int f = (int)(idx & 31);
    float v = agg1[idx] + ((f < H_OUT) ? b1[f] : 0.0f);
    agg1[idx] = fmaxf(v, 0.0f);          // padding cols stay exactly 0
}

__global__ void k_init_agg2(const float* __restrict__ h2, const float* __restrict__ dinv,
                            float* __restrict__ agg2, int N) {
    long long idx = (long long)blockIdx.x * blockDim.x + threadIdx.x;
    long long total = (long long)N * C_OUT;
    if (idx >= total) return;
    int node = (int)(idx / C_OUT);
    int f    = (int)(idx % C_OUT);
    float d = dinv[node];
    agg2[idx] = d * d * h2[(size_t)node * C_PAD + f];
}

__global__ void k_scatter2(const float* __restrict__ h2, const int* __restrict__ src,
                           const int* __restrict__ dst, const float* __restrict__ norm,
                           float* agg2, long long total) {
    long long idx = (long long)blockIdx.x * blockDim.x + threadIdx.x;
    if (idx >= total) return;
    int e = (int)(idx / C_OUT);
    int f = (int)(idx % C_OUT);
    float v = norm[e] * h2[(size_t)src[e] * C_PAD + f];
    atomAddF(&agg2[(size_t)dst[e] * C_OUT + f], v);
}

// ---------------- bias + log_softmax, one wave per node (C=40 -> lanes cover f and f+32) ----
__global__ void __launch_bounds__(256)
k_logsoftmax(const float* __restrict__ agg2, const float* __restrict__ b2,
             float* __restrict__ out, int N) {
    int lane = threadIdx.x & 31;
    int node = blockIdx.x * 8 + (threadIdx.x >> 5);
    if (node >= N) return;
    const float* row = agg2 + (size_t)node * C_OUT;
    float z1 = row[lane] + b2[lane];
    float z2 = (lane < 8) ? (row[32 + lane] + b2[32 + lane]) : -INFINITY;
    float mx = fmaxf(z1, z2);
    for (int off = 16; off > 0; off >>= 1) mx = fmaxf(mx, __shfl_xor(mx, off, 32));
    float s = expf(z1 - mx) + ((lane < 8) ? expf(z2 - mx) : 0.0f);
    for (int off = 16; off > 0; off >>= 1) s += __shfl_xor(s, off, 32);
    float ls = logf(s);
    float* orow = out + (size_t)node * C_OUT;
    orow[lane] = z1 - mx - ls;
    if (lane < 8) orow[32 + lane] = z2 - mx - ls;
}

// ---------------- host ----------------
extern "C" void kernel_launch(void* const* d_in, const int* in_sizes, int n_in,
                              void* d_out, int out_size, void* d_ws, size_t ws_size,
                              hipStream_t stream) {
    const float* x  = (const float*)d_in[0];
    const int*   ei = (const int*)d_in[1];     // [2,E] row-major
    const float* ew = (const float*)d_in[2];
    const float* W1 = (const float*)d_in[3];
    const float* b1 = (const float*)d_in[4];
    const float* W2 = (const float*)d_in[5];
    const float* b2 = (const float*)d_in[6];
    float* out = (float*)d_out;

    const int N = in_sizes[0] / F_IN;
    const int E = in_sizes[2];
    const int* src = ei;
    const int* dst = ei + E;

    // carve workspace (floats)
    float* ws = (float*)d_ws;
    size_t o = 0;
    float* deg  = ws + o; o += ((size_t)N + 63) & ~(size_t)63;   // deg -> dinv in place
    float* norm = ws + o; o += ((size_t)E + 63) & ~(size_t)63;
    float* h1   = ws + o; o += (size_t)N * H_PAD;
    float* agg1 = ws + o; o += (size_t)N * H_PAD;                // reused as h1r (ReLU in place)
    float* h2   = ws + o; o += (size_t)N * C_PAD;
    float* agg2 = ws + o; o += (size_t)N * C_OUT;

    const int T = 256;
    k_init_deg   <<<(N + T - 1) / T, T, 0, stream>>>(deg, N);
    k_deg_scatter<<<(E + T - 1) / T, T, 0, stream>>>(dst, ew, deg, E);
    k_rsqrt      <<<(N + T - 1) / T, T, 0, stream>>>(deg, N);
    k_norm       <<<(E + T - 1) / T, T, 0, stream>>>(src, dst, ew, deg, norm, E);

    k_gemm1<<<(N + 127) / 128, 128, 0, stream>>>(x, W1, h1, N);

    long long t1 = (long long)N * H_PAD;
    k_init_agg1<<<(unsigned)((t1 + T - 1) / T), T, 0, stream>>>(h1, deg, agg1, t1);
    long long s1 = (long long)E * 32;     // 32 lanes per edge, f<25 active
    k_scatter1 <<<(unsigned)((s1 + T - 1) / T), T, 0, stream>>>(h1, src, dst, norm, agg1, E);
    k_bias_relu<<<(unsigned)((t1 + T - 1) / T), T, 0, stream>>>(agg1, b1, t1);

    k_gemm2<<<(N + 127) / 128, 128, 0, stream>>>(agg1, W2, h2, N);

    long long t2 = (long long)N * C_OUT;
    k_init_agg2<<<(unsigned)((t2 + T - 1) / T), T, 0, stream>>>(h2, deg, agg2, N);
    long long s2 = (long long)E * C_OUT;
    k_scatter2 <<<(unsigned)((s2 + T - 1) / T), T, 0, stream>>>(h2, src, dst, norm, agg2, s2);

    k_logsoftmax<<<(N + 7) / 8, 256, 0, stream>>>(agg2, b2, out, N);
}